// PositionalEmbeddings2DWithBilinearInterpolation_16896401342851
// MI455X (gfx1250) — compile-verified
//
#include <hip/hip_runtime.h>
#include <hip/hip_bf16.h>
#include <stdint.h>

// 2D bilinear gather of positional embeddings.
//   positions : (8,256,256,2) f32  -> 524288 points
//   embeddings: (64,64,128)   f32  -> 2 MB table (L2-resident)
//   out       : (8,256,256,128) f32 -> 256 MB streamed (non-temporal)
//
// Mapping: one wave32 per point; lane l owns features [4l, 4l+4) as a float4.
// Each corner gather is one global_load_b128 per lane (512 B contiguous per
// wave); the output store is one NT global_store_b128 per lane.

typedef float v4f __attribute__((ext_vector_type(4)));

#define GXY  64
#define NF   128
#define ROW4 (NF / 4)          // 32 float4 per embedding row == one wave32
#define PTS_PER_BLOCK 8        // 256 threads = 8 waves = 8 points

__global__ __launch_bounds__(256) void
bilerp_gather_kernel(const float* __restrict__ positions,
                     const float* __restrict__ embeddings,
                     float* __restrict__ out,
                     long long npts)
{
    const int wave = threadIdx.x >> 5;
    const int lane = threadIdx.x & 31;
    const long long p0 = (long long)blockIdx.x * PTS_PER_BLOCK;

    __shared__ float sPos[PTS_PER_BLOCK * 2];

    // Stage this block's 8 position pairs (64 B) into LDS via the CDNA5
    // async global->LDS path (tracked by ASYNCcnt, no VGPR round trip).
    if (threadIdx.x < PTS_PER_BLOCK && (p0 + threadIdx.x) < npts) {
        const float* gp = positions + (p0 + threadIdx.x) * 2;
        // Low 32 bits of the flat shared-memory address are the LDS offset.
        unsigned ldsOff = (unsigned)(uintptr_t)&sPos[threadIdx.x * 2];
        asm volatile("global_load_async_to_lds_b64 %0, %1, off"
                     :: "v"(ldsOff), "v"(gp) : "memory");
    }
    // Pull the next block's position cacheline toward L2 (global_prefetch_b8).
    if (threadIdx.x == 0) {
        __builtin_prefetch(positions + (p0 + PTS_PER_BLOCK) * 2, 0, 1);
    }
    asm volatile("s_wait_asynccnt 0" ::: "memory");
    __syncthreads();

    const long long p = p0 + wave;
    if (p >= npts) return;

    // positions[...,0] indexes grid dim 0 (x), [...,1] indexes dim 1 (y).
    // SPATIAL_ORIGIN=(0,0), SPATIAL_SIZE=(1,1) -> coord * 64.
    const float px = sPos[wave * 2 + 0];
    const float py = sPos[wave * 2 + 1];

    const float x   = px * (float)GXY;
    const float y   = py * (float)GXY;
    const float fx0 = floorf(x);
    const float fy0 = floorf(y);
    const int   x0  = (int)fx0;
    const int   y0  = (int)fy0;
    const float dx  = x - fx0;
    const float dy  = y - fy0;

    const int x0c = min(max(x0,     0), GXY - 1);
    const int y0c = min(max(y0,     0), GXY - 1);
    const int x1  = min(max(x0 + 1, 0), GXY - 1);
    const int y1  = min(max(y0 + 1, 0), GXY - 1);

    // Four corner gathers: regular-temporal 128-bit loads so the 2 MB table
    // stays hot in WGP$/L2 (the NT output stream below avoids evicting it).
    const v4f* __restrict__ E = (const v4f*)embeddings;
    const v4f f00 = E[(x0c * GXY + y0c) * ROW4 + lane];  // E[x0c, y0c]
    const v4f f01 = E[(x0c * GXY + y1 ) * ROW4 + lane];  // E[x0c, y1 ]
    const v4f f10 = E[(x1  * GXY + y0c) * ROW4 + lane];  // E[x1 , y0c]
    const v4f f11 = E[(x1  * GXY + y1 ) * ROW4 + lane];  // E[x1 , y1 ]

    const float omdx = 1.0f - dx;
    const float omdy = 1.0f - dy;
    const v4f fxy0 = f00 * omdx + f10 * dx;   // lerp along x at y0
    const v4f fxy1 = f01 * omdx + f11 * dx;   // lerp along x at y1
    const v4f r    = fxy0 * omdy + fxy1 * dy; // lerp along y

    // Streaming (non-temporal) store: 256 MB write-once must not thrash the
    // 192 MB L2 that is serving ~1 GB of embedding-gather reuse.
    __builtin_nontemporal_store(r, (v4f*)out + p * (long long)ROW4 + lane);
}

extern "C" void kernel_launch(void* const* d_in, const int* in_sizes, int n_in,
                              void* d_out, int out_size, void* d_ws, size_t ws_size,
                              hipStream_t stream)
{
    const float* positions  = (const float*)d_in[0];
    const float* embeddings = (const float*)d_in[1];
    float* out = (float*)d_out;

    const long long npts = (long long)in_sizes[0] / 2;   // 524288
    const int blocks = (int)((npts + PTS_PER_BLOCK - 1) / PTS_PER_BLOCK);

    bilerp_gather_kernel<<<dim3(blocks), dim3(256), 0, stream>>>(
        positions, embeddings, out, npts);
}